// GraphGAT_88012469829906
// MI455X (gfx1250) — compile-verified
//
#include <hip/hip_runtime.h>
#include <math.h>

#define N_NODES   50000
#define N_EDGES   1600000
#define D_IN      512
#define D_HID     256
#define NEG_SLOPE 0.2f

typedef __attribute__((ext_vector_type(2))) float v2f;
typedef __attribute__((ext_vector_type(8))) float v8f;

// ---------------------------------------------------------------------------
// GEMM: H[rows x dout] = X[rows x din] @ W[din x dout], f32, via
// V_WMMA_F32_16X16X4_F32. Register blocking: each wave owns a 64x32 C strip
// (4 row-tiles x 2 col-tiles = 8 C fragments, 64 VGPRs of accumulator), so a
// K-step is 4xb64 A loads + 4xb32 B loads feeding 8 WMMAs (1:1 ratio).
// Block = 4 waves covering 64 rows x 128 cols. Tail rows handled by clamping
// A loads (duplicate last row) and predicating stores only -> EXEC stays
// all-ones for every WMMA as required.
// ---------------------------------------------------------------------------
__global__ __launch_bounds__(128)
void gat_gemm_wmma(const float* __restrict__ X, const float* __restrict__ W,
                   float* __restrict__ H, int rows, int din, int dout) {
  const int wave = threadIdx.x >> 5;
  const int lane = threadIdx.x & 31;
  const int m = lane & 15;   // A: row within tile, B/C: col within tile
  const int g = lane >> 4;   // half-wave select

  const int row0 = blockIdx.x * 64;
  const int col0 = blockIdx.y * 128 + wave * 32;

  const float* arow[4];
  #pragma unroll
  for (int i = 0; i < 4; ++i) {
    int r = row0 + 16 * i + m;
    r = (r < rows) ? r : (rows - 1);          // clamp for tail block
    arow[i] = X + (size_t)r * (size_t)din;
  }
  const float* bcol0 = W + col0 + m;
  const float* bcol1 = W + col0 + 16 + m;

  v8f acc[4][2] = {};

  #pragma unroll 2
  for (int k0 = 0; k0 < din; k0 += 4) {
    const int ka = k0 + (g << 1);
    // A 16x4 f32: lane (m + 16g) holds (M=m, K=2g) and (M=m, K=2g+1)
    v2f a[4];
    #pragma unroll
    for (int i = 0; i < 4; ++i) {
      a[i].x = arow[i][ka];
      a[i].y = arow[i][ka + 1];
    }
    // B 4x16 f32: lane (n + 16g) holds (K=2g, n) and (K=2g+1, n)
    const size_t boff = (size_t)ka * (size_t)dout;
    v2f b[2];
    b[0].x = bcol0[boff]; b[0].y = bcol0[boff + dout];
    b[1].x = bcol1[boff]; b[1].y = bcol1[boff + dout];

    #pragma unroll
    for (int i = 0; i < 4; ++i) {
      acc[i][0] = __builtin_amdgcn_wmma_f32_16x16x4_f32(false, a[i], false, b[0],
                                                        (short)0, acc[i][0],
                                                        false, false);
      acc[i][1] = __builtin_amdgcn_wmma_f32_16x16x4_f32(false, a[i], false, b[1],
                                                        (short)0, acc[i][1],
                                                        false, false);
    }
  }

  // C 16x16 f32: VGPR j, lanes 0-15 -> M=j, lanes 16-31 -> M=8+j; N = lane&15
  #pragma unroll
  for (int i = 0; i < 4; ++i) {
    const int rbase = row0 + 16 * i + (g << 3);
    #pragma unroll
    for (int j = 0; j < 8; ++j) {
      const int r = rbase + j;
      if (r < rows) {
        float* out = H + (size_t)r * (size_t)dout + col0 + m;
        out[0]  = acc[i][0][j];
        out[16] = acc[i][1][j];
      }
    }
  }
}

// ---------------------------------------------------------------------------
// Per-node attention logits: as_n = h . a_src, ad_n = h . a_dst.
// One wave per node (D_HID = 256 -> 8 elems/lane), wave32 shuffle reduction.
// ---------------------------------------------------------------------------
__global__ __launch_bounds__(256)
void gat_scores(const float* __restrict__ H, const float* __restrict__ a_src,
                const float* __restrict__ a_dst, float* __restrict__ as_n,
                float* __restrict__ ad_n) {
  const int node = blockIdx.x * 8 + (threadIdx.x >> 5);
  if (node >= N_NODES) return;
  const int lane = threadIdx.x & 31;
  const float* h = H + (size_t)node * D_HID;
  float vs = 0.f, vd = 0.f;
  #pragma unroll
  for (int j = 0; j < D_HID / 32; ++j) {
    const int c = lane + 32 * j;
    const float x = h[c];
    vs += x * a_src[c];
    vd += x * a_dst[c];
  }
  #pragma unroll
  for (int off = 16; off > 0; off >>= 1) {
    vs += __shfl_xor(vs, off, 32);
    vd += __shfl_xor(vd, off, 32);
  }
  if (lane == 0) { as_n[node] = vs; ad_n[node] = vd; }
}

// ---------------------------------------------------------------------------
// Per-layer state init: acc = 0, segment-max = -inf, segment-sum = 0.
// ---------------------------------------------------------------------------
__global__ void gat_init(float* __restrict__ acc, float* __restrict__ m,
                         float* __restrict__ s) {
  const size_t i = (size_t)blockIdx.x * blockDim.x + threadIdx.x;
  if (i < (size_t)N_NODES * D_HID) acc[i] = 0.f;
  if (i < N_NODES) { m[i] = -__builtin_inff(); s[i] = 0.f; }
}

__device__ __forceinline__ void edge_nodes(const int* __restrict__ ei, int idx,
                                           int& src, int& dst) {
  if (idx < N_EDGES) { src = ei[idx]; dst = ei[N_EDGES + idx]; }
  else               { src = dst = idx - N_EDGES; }  // self loop
}

__device__ __forceinline__ float lrelu(float z) {
  return z > 0.f ? z : NEG_SLOPE * z;
}

// float atomic-max via signed-int max (>=0) / unsigned-int min (<0)
__device__ __forceinline__ void atomicMaxF(float* addr, float v) {
  if (v >= 0.f) atomicMax((int*)addr, __float_as_int(v));
  else          atomicMin((unsigned int*)addr, __float_as_uint(v));
}

// ---------------------------------------------------------------------------
// Pass 1 over edges: segment max of leaky-relu logits into m[dst].
// ---------------------------------------------------------------------------
__global__ __launch_bounds__(256)
void gat_edge_max(const int* __restrict__ ei, const float* __restrict__ as_n,
                  const float* __restrict__ ad_n, float* __restrict__ m) {
  const int idx = blockIdx.x * blockDim.x + threadIdx.x;
  if (idx >= N_EDGES + N_NODES) return;
  int src, dst;
  edge_nodes(ei, idx, src, dst);
  atomicMaxF(&m[dst], lrelu(as_n[src] + ad_n[dst]));
}

// ---------------------------------------------------------------------------
// Pass 2 over edges: s[dst] += exp(e - m[dst]).
// ---------------------------------------------------------------------------
__global__ __launch_bounds__(256)
void gat_edge_sum(const int* __restrict__ ei, const float* __restrict__ as_n,
                  const float* __restrict__ ad_n, const float* __restrict__ m,
                  float* __restrict__ s) {
  const int idx = blockIdx.x * blockDim.x + threadIdx.x;
  if (idx >= N_EDGES + N_NODES) return;
  int src, dst;
  edge_nodes(ei, idx, src, dst);
  const float e = lrelu(as_n[src] + ad_n[dst]);
  atomicAdd(&s[dst], __expf(e - m[dst]));
}

// ---------------------------------------------------------------------------
// Pass 3 over edges: acc[dst] += alpha * h[src]. One wave per edge; gather of
// h[src] uses 2x global_load_b128 per lane; scatter is 8x f32 atomics into
// the L2-resident (192 MB L2) accumulator.
// ---------------------------------------------------------------------------
__global__ __launch_bounds__(256)
void gat_aggregate(const int* __restrict__ ei, const float* __restrict__ H,
                   const float* __restrict__ as_n, const float* __restrict__ ad_n,
                   const float* __restrict__ m, const float* __restrict__ s,
                   float* __restrict__ acc) {
  const int edge = blockIdx.x * 8 + (threadIdx.x >> 5);
  if (edge >= N_EDGES + N_NODES) return;
  const int lane = threadIdx.x & 31;
  int src, dst;
  edge_nodes(ei, edge, src, dst);
  const float e = lrelu(as_n[src] + ad_n[dst]);
  const float alpha = __expf(e - m[dst]) / (s[dst] + 1e-16f);
  const float4* hs4 = (const float4*)(H + (size_t)src * D_HID);
  float* ap = acc + (size_t)dst * D_HID;
  #pragma unroll
  for (int j = 0; j < D_HID / 128; ++j) {       // 2 iterations: 64 float4/row
    const int c4 = lane + 32 * j;
    const float4 v = hs4[c4];
    float* a4 = ap + c4 * 4;
    atomicAdd(a4 + 0, alpha * v.x);
    atomicAdd(a4 + 1, alpha * v.y);
    atomicAdd(a4 + 2, alpha * v.z);
    atomicAdd(a4 + 3, alpha * v.w);
  }
}

// ---------------------------------------------------------------------------
// Finalize: out = [resid +] relu(acc + b)
// ---------------------------------------------------------------------------
__global__ __launch_bounds__(256)
void gat_finalize(const float* __restrict__ acc, const float* __restrict__ b,
                  const float* __restrict__ resid, float* __restrict__ out,
                  int add_resid) {
  const size_t i = (size_t)blockIdx.x * blockDim.x + threadIdx.x;
  if (i >= (size_t)N_NODES * D_HID) return;
  const int col = (int)(i & (D_HID - 1));
  float v = fmaxf(acc[i] + b[col], 0.f);
  if (add_resid) v += resid[i];
  out[i] = v;
}

// ---------------------------------------------------------------------------
extern "C" void kernel_launch(void* const* d_in, const int* in_sizes, int n_in,
                              void* d_out, int out_size, void* d_ws, size_t ws_size,
                              hipStream_t stream) {
  (void)in_sizes; (void)n_in; (void)out_size; (void)ws_size;

  const float* x  = (const float*)d_in[0];
  const int*   ei = (const int*)d_in[1];
  const float* W[3]   = {(const float*)d_in[2], (const float*)d_in[6],  (const float*)d_in[10]};
  const float* asv[3] = {(const float*)d_in[3], (const float*)d_in[7],  (const float*)d_in[11]};
  const float* adv[3] = {(const float*)d_in[4], (const float*)d_in[8],  (const float*)d_in[12]};
  const float* bv[3]  = {(const float*)d_in[5], (const float*)d_in[9],  (const float*)d_in[13]};

  // Workspace layout (floats): hbuf | acc | xin | as_n | ad_n | m | s
  float* ws   = (float*)d_ws;
  const size_t nd = (size_t)N_NODES * D_HID;
  float* hbuf = ws;
  float* acc  = hbuf + nd;
  float* xin  = acc + nd;
  float* as_n = xin + nd;
  float* ad_n = as_n + N_NODES;
  float* mbuf = ad_n + N_NODES;
  float* sbuf = mbuf + N_NODES;

  const int ETOT = N_EDGES + N_NODES;
  const dim3 gemmGrid((N_NODES + 63) / 64, D_HID / 128);  // 782 x 2
  const dim3 gemmBlock(128);
  const int ndBlocks       = (int)((nd + 255) / 256);
  const int scoreBlocks    = (N_NODES + 7) / 8;
  const int edgeBlocks     = (ETOT + 255) / 256;
  const int edgeWaveBlocks = (ETOT + 7) / 8;

  for (int layer = 0; layer < 3; ++layer) {
    const float* xin_l = (layer == 0) ? x : xin;
    const int din = (layer == 0) ? D_IN : D_HID;

    gat_gemm_wmma<<<gemmGrid, gemmBlock, 0, stream>>>(xin_l, W[layer], hbuf,
                                                      N_NODES, din, D_HID);
    gat_scores<<<scoreBlocks, 256, 0, stream>>>(hbuf, asv[layer], adv[layer], as_n, ad_n);
    gat_init<<<ndBlocks, 256, 0, stream>>>(acc, mbuf, sbuf);
    gat_edge_max<<<edgeBlocks, 256, 0, stream>>>(ei, as_n, ad_n, mbuf);
    gat_edge_sum<<<edgeBlocks, 256, 0, stream>>>(ei, as_n, ad_n, mbuf, sbuf);
    gat_aggregate<<<edgeWaveBlocks, 256, 0, stream>>>(ei, hbuf, as_n, ad_n, mbuf, sbuf, acc);

    float* outp = (layer == 2) ? (float*)d_out : xin;
    gat_finalize<<<ndBlocks, 256, 0, stream>>>(acc, bv[layer], xin, outp, layer > 0 ? 1 : 0);
  }
}